// EPSesPlusLinear_72413148610569
// MI455X (gfx1250) — compile-verified
//
#include <hip/hip_runtime.h>
#include <hip/hip_bf16.h>
#include <stdint.h>

typedef __attribute__((ext_vector_type(2))) float v2f;
typedef __attribute__((ext_vector_type(8))) float v8f;

// ---------------------------------------------------------------------------
// Threefry-2x32 (exact JAX semantics) for the dropconnect masks
// ---------------------------------------------------------------------------
__device__ __forceinline__ uint32_t rotl32(uint32_t x, int d) {
    return (x << d) | (x >> (32 - d));
}

__device__ __forceinline__ void threefry2x32(uint32_t k0, uint32_t k1,
                                             uint32_t x0, uint32_t x1,
                                             uint32_t* o0, uint32_t* o1) {
    uint32_t ks[3] = {k0, k1, k0 ^ k1 ^ 0x1BD11BDAu};
    const int rotA[4] = {13, 15, 26, 6};
    const int rotB[4] = {17, 29, 16, 24};
    x0 += ks[0];
    x1 += ks[1];
#pragma unroll
    for (int g = 0; g < 5; ++g) {
        const int* r = (g & 1) ? rotB : rotA;
#pragma unroll
        for (int j = 0; j < 4; ++j) {
            x0 += x1;
            x1 = rotl32(x1, r[j]);
            x1 ^= x0;
        }
        x0 += ks[(g + 1) % 3];
        x1 += ks[(g + 2) % 3] + (uint32_t)(g + 1);
    }
    *o0 = x0;
    *o1 = x1;
}

__device__ __forceinline__ float bits_to_unit_float(uint32_t b) {
    uint32_t u = (b >> 9) | 0x3f800000u;
    return __uint_as_float(u) - 1.0f;
}

// masked value = (u < 0.8) ? core * 1.25 : 0
__device__ __forceinline__ float apply_mask(float core, uint32_t bits) {
    return (bits_to_unit_float(bits) < 0.8f) ? core * 1.25f : 0.0f;
}

// B layout: lane-pair-interleaved so each lane reads one aligned float2/chunk.
// element (k, n) -> B[((k>>2)*2 + ((k>>1)&1))*32 + n*2 + (k&1)]
__device__ __forceinline__ int b_index(int k, int n) {
    return (((k >> 2) * 2 + ((k >> 1) & 1)) * 32) + n * 2 + (k & 1);
}

// ---------------------------------------------------------------------------
// Prep kernels: build masked, transposed, zero-padded B matrices
// ---------------------------------------------------------------------------
// eps0: (6,512)  -> B1, n = 3072, half = 1536, fold_in(key42, 0)
__global__ void prep_eps0(const float* __restrict__ eps0, float* __restrict__ B1) {
    int j = blockIdx.x * blockDim.x + threadIdx.x;
    const int half = 1536;
    if (j >= half) return;
    uint32_t k0, k1;
    threefry2x32(0u, 42u, 0u, 0u, &k0, &k1);  // fold_in(key, 0)
    uint32_t b0, b1;
    threefry2x32(k0, k1, (uint32_t)j, (uint32_t)(j + half), &b0, &b1);
    int e0 = j, e1 = j + half;
    B1[b_index(e0 & 511, e0 >> 9)] = apply_mask(eps0[e0], b0);
    B1[b_index(e1 & 511, e1 >> 9)] = apply_mask(eps0[e1], b1);
}

// eps1: (10,1296) -> B2, n = 12960, half = 6480, fold_in(key42, 1)
__global__ void prep_eps1(const float* __restrict__ eps1, float* __restrict__ B2) {
    int j = blockIdx.x * blockDim.x + threadIdx.x;
    const int half = 6480;
    if (j >= half) return;
    uint32_t k0, k1;
    threefry2x32(0u, 42u, 0u, 1u, &k0, &k1);  // fold_in(key, 1)
    uint32_t b0, b1;
    threefry2x32(k0, k1, (uint32_t)j, (uint32_t)(j + half), &b0, &b1);
    int e0 = j, e1 = j + half;
    B2[b_index(e0 % 1296, e0 / 1296)] = apply_mask(eps1[e0], b0);
    B2[b_index(e1 % 1296, e1 / 1296)] = apply_mask(eps1[e1], b1);
}

// W: (10,6250) -> B3 (no mask); k rows 6250,6251 stay zero (memset)
__global__ void prep_W(const float* __restrict__ W, float* __restrict__ B3) {
    int e = blockIdx.x * blockDim.x + threadIdx.x;
    if (e >= 62500) return;
    int n = e / 6250;
    int k = e - n * 6250;
    B3[b_index(k, n)] = W[e];
}

// ---------------------------------------------------------------------------
// f32 WMMA helper: D = A(16x4) * B(4x16) + C
// ---------------------------------------------------------------------------
__device__ __forceinline__ v8f wmma4(v2f a, v2f b, v8f c) {
    return __builtin_amdgcn_wmma_f32_16x16x4_f32(false, a, false, b, (short)0, c,
                                                 false, false);
}

// ---------------------------------------------------------------------------
// Stage 1: x(128,28,28,2), 3x3 windows -> K=512 tensor product, x B1 -> h1
// 86528 positions, 16 per wave -> 5408 waves = 676 blocks x 8 waves
// ---------------------------------------------------------------------------
__global__ void __launch_bounds__(256) stage1(const float* __restrict__ x,
                                              const float* __restrict__ B1,
                                              float* __restrict__ h1) {
    const int lane = threadIdx.x & 31;
    const int wave = (blockIdx.x * blockDim.x + threadIdx.x) >> 5;
    const int row = lane & 15;     // M row within tile / B column
    const int hi = lane >> 4;      // K sub-slot select
    const int mbase = wave * 16;
    const int m = mbase + row;     // flat position (b,i,j), exact multiple of 16

    int jj = m % 26;
    int t = m / 26;
    int ii = t % 26;
    int bb = t / 26;

    // Load the 9 window pixels (2 channels each)
    float px[9][2];
    const float* xb = x + (((bb * 28 + ii) * 28) + jj) * 2;
#pragma unroll
    for (int di = 0; di < 3; ++di)
#pragma unroll
        for (int dj = 0; dj < 3; ++dj) {
            const float* p = xb + (di * 28 + dj) * 2;
            px[di * 3 + dj][0] = p[0];
            px[di * 3 + dj][1] = p[1];
        }

    // Lane-dependent pixel-7 factor (single select), pixel-8 pair
    const float f7 = hi ? px[7][1] : px[7][0];
    const float f80 = px[8][0];
    const float f81 = px[8][1];

    v8f acc = {};
    const float* bp = B1 + hi * 32 + row * 2;  // one float2 per chunk, +64/chunk
    // k = c0*256 + c1*128 + ... + c6*4 + c7*2 + c8 ; lane covers c7=hi, c8={0,1}
#pragma unroll
    for (int c0 = 0; c0 < 2; ++c0) { float p0 = px[0][c0];
#pragma unroll
    for (int c1 = 0; c1 < 2; ++c1) { float p1 = p0 * px[1][c1];
#pragma unroll
    for (int c2 = 0; c2 < 2; ++c2) { float p2 = p1 * px[2][c2];
#pragma unroll
    for (int c3 = 0; c3 < 2; ++c3) { float p3 = p2 * px[3][c3];
#pragma unroll
    for (int c4 = 0; c4 < 2; ++c4) { float p4 = p3 * px[4][c4];
#pragma unroll
    for (int c5 = 0; c5 < 2; ++c5) { float p5 = p4 * px[5][c5];
#pragma unroll
    for (int c6 = 0; c6 < 2; ++c6) { float p6 = p5 * px[6][c6];
        float q = p6 * f7;
        v2f a;
        a[0] = q * f80;
        a[1] = q * f81;
        v2f b = *(const v2f*)bp;
        bp += 64;
        acc = wmma4(a, b, acc);
    }}}}}}}

    // C layout: lane n=row, VGPR r -> M = r + 8*hi ; keep only 6 output cols
    if (row < 6) {
#pragma unroll
        for (int r = 0; r < 8; ++r) {
            int mm = mbase + r + 8 * hi;
            h1[mm * 6 + row] = acc[r];
        }
    }
}

// ---------------------------------------------------------------------------
// Stage 2: h1(128,26,26,6), 2x2 windows -> K=1296 tensor product, x B2 -> flat
// 80000 positions -> 5000 waves = 625 blocks x 8 waves
// ---------------------------------------------------------------------------
__global__ void __launch_bounds__(256) stage2(const float* __restrict__ h1,
                                              const float* __restrict__ B2,
                                              float* __restrict__ flat) {
    const int lane = threadIdx.x & 31;
    const int wave = (blockIdx.x * blockDim.x + threadIdx.x) >> 5;
    const int row = lane & 15;
    const int hi = lane >> 4;
    const int mbase = wave * 16;
    const int m = mbase + row;  // flat position over (128,25,25)

    int jj = m % 25;
    int t = m / 25;
    int ii = t % 25;
    int bb = t / 25;

    // 4 window pixels x 6 channels
    float hp[4][6];
#pragma unroll
    for (int di = 0; di < 2; ++di)
#pragma unroll
        for (int dj = 0; dj < 2; ++dj) {
            const float* p = h1 + (((bb * 26 + ii + di) * 26) + jj + dj) * 6;
#pragma unroll
            for (int c = 0; c < 6; ++c) hp[di * 2 + dj][c] = p[c];
        }

    // Per-lane (c2,c3) pair products for the 9 aligned chunks of the inner
    // 36-value block: compile-time indices, single select per value on hi.
    float q0[9], q1[9];
#pragma unroll
    for (int u = 0; u < 9; ++u) {
        const int ka = u * 4;      // hi=0, slot 0
        const int kb = u * 4 + 1;  // hi=0, slot 1
        const int kc = u * 4 + 2;  // hi=1, slot 0
        const int kd = u * 4 + 3;  // hi=1, slot 1
        float va = hp[2][ka / 6] * hp[3][ka % 6];
        float vb = hp[2][kb / 6] * hp[3][kb % 6];
        float vc = hp[2][kc / 6] * hp[3][kc % 6];
        float vd = hp[2][kd / 6] * hp[3][kd % 6];
        q0[u] = hi ? vc : va;
        q1[u] = hi ? vd : vb;
    }

    v8f acc = {};
    const float* bp = B2 + hi * 32 + row * 2;  // one float2 per chunk, +64/chunk
    // k = c0*216 + c1*36 + c2*6 + c3 ; 36-value inner block = 9 aligned chunks
    for (int c0 = 0; c0 < 6; ++c0) {
        for (int c1 = 0; c1 < 6; ++c1) {
            float pre = hp[0][c0] * hp[1][c1];
#pragma unroll
            for (int u = 0; u < 9; ++u) {
                v2f a;
                a[0] = pre * q0[u];
                a[1] = pre * q1[u];
                v2f b = *(const v2f*)bp;
                bp += 64;
                acc = wmma4(a, b, acc);
            }
        }
    }

    if (row < 10) {
#pragma unroll
        for (int r = 0; r < 8; ++r) {
            int mm = mbase + r + 8 * hi;
            flat[mm * 10 + row] = acc[r];
        }
    }
}

// ---------------------------------------------------------------------------
// Stage 3: flat(128,6250) x B3 (= W^T zero-padded to 6252) + bias -> out(128,10)
// 8 waves, one block. K chunks: 1562 full + 1 guarded tail = 1563.
// Loop blocked 390x4 + 2 to keep code size bounded (no full unroll).
// ---------------------------------------------------------------------------
__global__ void __launch_bounds__(256) stage3(const float* __restrict__ flat,
                                              const float* __restrict__ B3,
                                              const float* __restrict__ bias,
                                              float* __restrict__ out) {
    const int lane = threadIdx.x & 31;
    const int wave = threadIdx.x >> 5;
    const int row = lane & 15;
    const int hi = lane >> 4;
    const int mbase = wave * 16;  // batch rows 0..127
    const float* arow = flat + (mbase + row) * 6250 + 2 * hi;

    v8f acc = {};
    const float* bp = B3 + hi * 32 + row * 2;  // one float2 per chunk, +64/chunk
    // Main body: 390 blocks of 4 chunks (k < 6240); keep outer loop rolled.
#pragma unroll 1
    for (int t0 = 0; t0 < 390; ++t0) {
#pragma unroll
        for (int s = 0; s < 4; ++s) {
            v2f a = *(const v2f*)(arow + (t0 * 4 + s) * 4);
            v2f b = *(const v2f*)bp;
            bp += 64;
            acc = wmma4(a, b, acc);
        }
    }
    // Chunks 1560, 1561 (k = 6240..6247, fully in range)
#pragma unroll
    for (int s = 0; s < 2; ++s) {
        v2f a = *(const v2f*)(arow + (1560 + s) * 4);
        v2f b = *(const v2f*)bp;
        bp += 64;
        acc = wmma4(a, b, acc);
    }
    // Tail chunk 1562: k = 6248..6251; hi=1 lanes are fully out of range
    {
        int k0 = 1562 * 4 + 2 * hi;
        v2f a;
        const float* arow0 = flat + (mbase + row) * 6250;
        a[0] = (k0 < 6250) ? arow0[k0] : 0.0f;
        a[1] = (k0 + 1 < 6250) ? arow0[k0 + 1] : 0.0f;
        v2f b = *(const v2f*)bp;
        acc = wmma4(a, b, acc);
    }

    if (row < 10) {
        float bi = bias[row];
#pragma unroll
        for (int r = 0; r < 8; ++r) {
            int mm = mbase + r + 8 * hi;
            out[mm * 10 + row] = acc[r] + bi;
        }
    }
}

// ---------------------------------------------------------------------------
// Workspace layout (floats):
//   B1: [0, 8192)          512*16
//   B2: [8192, 28928)      1296*16
//   B3: [28928, 128960)    6252*16
//   h1: [128960, 648128)   86528*6
//   flat: [648128, 1448128) 80000*10
// total ~5.8 MB
// ---------------------------------------------------------------------------
extern "C" void kernel_launch(void* const* d_in, const int* in_sizes, int n_in,
                              void* d_out, int out_size, void* d_ws, size_t ws_size,
                              hipStream_t stream) {
    const float* x = (const float*)d_in[0];
    const float* eps0 = (const float*)d_in[1];
    const float* eps1 = (const float*)d_in[2];
    const float* W = (const float*)d_in[3];
    const float* bias = (const float*)d_in[4];
    float* out = (float*)d_out;
    float* ws = (float*)d_ws;

    float* B1 = ws;
    float* B2 = ws + 8192;
    float* B3 = ws + 28928;
    float* h1 = ws + 128960;
    float* flat = ws + 648128;

    // Zero all B pads (contiguous region), capture-safe
    hipMemsetAsync(B1, 0, (size_t)128960 * sizeof(float), stream);

    prep_eps0<<<(1536 + 255) / 256, 256, 0, stream>>>(eps0, B1);
    prep_eps1<<<(6480 + 255) / 256, 256, 0, stream>>>(eps1, B2);
    prep_W<<<(62500 + 255) / 256, 256, 0, stream>>>(W, B3);

    stage1<<<676, 256, 0, stream>>>(x, B1, h1);     // 5408 waves * 16 rows = 86528
    stage2<<<625, 256, 0, stream>>>(h1, B2, flat);  // 5000 waves * 16 rows = 80000
    stage3<<<1, 256, 0, stream>>>(flat, B3, bias, out);  // 8 waves * 16 rows = 128
}